// SophisticatedETADPooling_43147241456199
// MI455X (gfx1250) — compile-verified
//
#include <hip/hip_runtime.h>

// ---------------------------------------------------------------------------
// SophisticatedETADPooling, collapsed form:
//   out[b,c,h,w] = c0 + (1/4096) * sum_{t} g[t] * sum_{h,w} x[b,c,t,h,w]
// g[32], c0 computed on-device from the small weight tensors (kernel A),
// heavy 128MB streaming reduction in kernel B1, 4MB broadcast in B2.
// ---------------------------------------------------------------------------

typedef __attribute__((ext_vector_type(2))) float v2f;
typedef __attribute__((ext_vector_type(8))) float v8f;

#define T_DIM 32
#define HW 4096            // 64*64
#define PER_BC 131072      // T*H*W floats per (b,c)
#define WS_G 0             // ws[0..31]  : g[t]
#define WS_C0 32           // ws[32]     : c0
#define WS_PART 33         // ws[33 + bc*4 + p] : 1024 partials

__device__ __forceinline__ float sigmoidf_(float x) {
  return 1.0f / (1.0f + expf(-x));
}

__device__ __forceinline__ float wave_sum32(float v) {
  // wave32 butterfly reduction; all lanes end with the total
  #pragma unroll
  for (int m = 16; m > 0; m >>= 1) v += __shfl_xor(v, m, 32);
  return v;
}

// ---------------------------------------------------------------------------
// Kernel A: one wave (32 lanes, lane == t). Computes g[32] and c0.
// Uses V_WMMA_F32_16X16X4_F32 for v = proj_w @ attn_value_flat (the only true
// matmul in the op). Trick: fold av into A and use B = all-ones, so
// D[m,n] = sum_k A[m,k] regardless of B's lane layout; every D column holds
// the matvec result.
// ---------------------------------------------------------------------------
__global__ void etad_weights_kernel(
    const float* __restrict__ conv_w,      // [32,1,3]
    const float* __restrict__ conv_b,      // [32]
    const float* __restrict__ scale_w,     // [4]
    const float* __restrict__ alpha_p,     // [1]
    const float* __restrict__ beta_p,      // [1]
    const float* __restrict__ t_imp,       // [32]
    const float* __restrict__ attn_value,  // [4,8] flat = 32
    const float* __restrict__ proj_w,      // [32,32]
    const float* __restrict__ proj_b,      // [32]
    const float* __restrict__ t_dyn,       // [32]
    const float* __restrict__ ln_gamma,    // [32]
    const float* __restrict__ ln_beta,     // [32]
    float* __restrict__ ws) {
  const int lane = threadIdx.x;  // 0..31 == t
  __shared__ float sv[T_DIM];    // v[t] = proj_w @ av (pre-bias)
  __shared__ float u[T_DIM];     // interp-adjoint scratch
  __shared__ float w2a[T_DIM];   // conv-adjoint scratch

  // channel-averaged conv kernel and bias (wave reductions, T == warpSize)
  const float bbar = wave_sum32(conv_b[lane]) * (1.0f / 32.0f);
  const float wb0  = wave_sum32(conv_w[lane * 3 + 0]) * (1.0f / 32.0f);
  const float wb1  = wave_sum32(conv_w[lane * 3 + 1]) * (1.0f / 32.0f);
  const float wb2  = wave_sum32(conv_w[lane * 3 + 2]) * (1.0f / 32.0f);

  // ---- WMMA: v = proj_w(32x32) @ av(32), two 16-row tiles, K chained x8 ----
  // A layout (f32 16x4): lanes 0-15 hold M=0..15 {K=0 in VGPR0, K=1 in VGPR1};
  // lanes 16-31 hold {K=2, K=3}.  (EXEC is all-1s: full wave, no divergence.)
  #pragma unroll
  for (int tile = 0; tile < 2; ++tile) {
    const int mrow = (tile << 4) + (lane & 15);
    const int koff = (lane >> 4) << 1;  // 0 or 2
    v8f c = {};
    #pragma unroll
    for (int k0 = 0; k0 < 32; k0 += 4) {
      const int k = k0 + koff;
      v2f a;
      a.x = proj_w[mrow * 32 + k]     * attn_value[k];
      a.y = proj_w[mrow * 32 + k + 1] * attn_value[k + 1];
      v2f bo;
      bo.x = 1.0f;
      bo.y = 1.0f;
      c = __builtin_amdgcn_wmma_f32_16x16x4_f32(
          /*neg_a=*/false, a, /*neg_b=*/false, bo,
          /*c_mod=*/(short)0, c, /*reuse_a=*/false, /*reuse_b=*/false);
    }
    // D columns are identical; lane n<16: VGPR r = D[M=r,*], lanes>=16: M=8+r
    if (lane == 0) {
      #pragma unroll
      for (int r = 0; r < 8; ++r) sv[(tile << 4) + r] = c[r];
    }
    if (lane == 16) {
      #pragma unroll
      for (int r = 0; r < 8; ++r) sv[(tile << 4) + 8 + r] = c[r];
    }
  }
  __syncthreads();

  const float vt = sv[lane] + proj_b[lane];  // constant "attention" vector

  // adaptive decay and LayerNorm(sigmoid(temporal_dynamics)) weights
  const float tf    = (float)lane;
  const float alpha = alpha_p[0];
  const float betad = beta_p[0];
  const float ad = expf(-alpha * tf) * sigmoidf_(betad * tf) * sigmoidf_(t_imp[lane]);

  const float tw   = sigmoidf_(t_dyn[lane]);
  const float mean = wave_sum32(tw) * (1.0f / 32.0f);
  const float dv   = tw - mean;
  const float var  = wave_sum32(dv * dv) * (1.0f / 32.0f);
  const float twn  = dv * rsqrtf(var + 1e-5f) * ln_gamma[lane] + ln_beta[lane];

  const float wf = ad * vt * twn;  // final per-t weight (pre multi-scale)

  // ---- g = sum_s sw_s * D_s^T C_lin^T I_s^T wf  (adjoint chain, lane==t) ----
  float g = 0.0f;
  for (int si = 0; si < 4; ++si) {
    const int s  = 1 << si;
    const int ts = T_DIM >> si;
    u[lane] = 0.0f;
    __syncthreads();
    // I^T (trilinear temporal interp adjoint, align_corners=False, clipped)
    const float tsf = (float)ts;
    float pos = (tf + 0.5f) * (tsf * 0.03125f) - 0.5f;
    pos = fminf(fmaxf(pos, 0.0f), tsf - 1.0f);
    const int i0 = (int)pos;  // pos >= 0 -> trunc == floor
    const int i1 = (i0 + 1 < ts) ? i0 + 1 : ts - 1;
    const float wq = pos - (float)i0;
    atomicAdd(&u[i0], wf * (1.0f - wq));
    atomicAdd(&u[i1], wf * wq);
    __syncthreads();
    // C_lin^T (zero-padded 3-tap conv adjoint; bias handled in c0)
    if (lane < ts) {
      const float uc = u[lane];
      const float um = (lane > 0) ? u[lane - 1] : 0.0f;
      const float up = (lane + 1 < ts) ? u[lane + 1] : 0.0f;
      w2a[lane] = wb0 * up + wb1 * uc + wb2 * um;
    }
    __syncthreads();
    // D_s^T (group-mean adjoint: spread coarse value / s)
    g += scale_w[si] * w2a[lane >> si] * (1.0f / (float)s);
    __syncthreads();
  }

  ws[WS_G + lane] = g;
  const float sum_wf = wave_sum32(wf);
  if (lane == 0) {
    const float sum_sw = scale_w[0] + scale_w[1] + scale_w[2] + scale_w[3];
    ws[WS_C0] = bbar * sum_sw * sum_wf;  // bias path dotted with wf
  }
}

// ---------------------------------------------------------------------------
// Kernel B1: weighted streaming reduction over x. grid = (256 bc, 4 chunks),
// 256 threads. Each block handles 8 t-planes (128KB contiguous) with float4
// (global_load_b128) loads; deterministic partials (no float atomics).
// ---------------------------------------------------------------------------
__global__ void etad_reduce_kernel(const float* __restrict__ x,
                                   float* __restrict__ ws) {
  const int bc  = blockIdx.x;
  const int p   = blockIdx.y;
  const int tid = threadIdx.x;

  __shared__ float gs[8];
  __shared__ float red[8];
  if (tid < 8) gs[tid] = ws[WS_G + p * 8 + tid];
  __syncthreads();

  const float4* __restrict__ xv =
      (const float4*)x + (size_t)bc * (PER_BC / 4) + (size_t)p * 8192;

  float acc = 0.0f;
  #pragma unroll 4
  for (int i = tid; i < 8192; i += 256) {  // 8192 float4 = 8 t-planes
    const float4 v = xv[i];
    acc = fmaf(gs[i >> 10], (v.x + v.y) + (v.z + v.w), acc);
  }

  acc = wave_sum32(acc);
  if ((tid & 31) == 0) red[tid >> 5] = acc;
  __syncthreads();
  if (tid == 0) {
    float s = 0.0f;
    #pragma unroll
    for (int w = 0; w < 8; ++w) s += red[w];
    ws[WS_PART + bc * 4 + p] = s;
  }
}

// ---------------------------------------------------------------------------
// Kernel B2: broadcast pooled value over H,W with b128 stores (4 MB).
// ---------------------------------------------------------------------------
__global__ void etad_broadcast_kernel(const float* __restrict__ ws,
                                      float4* __restrict__ out) {
  const int o = blockIdx.x * 256 + threadIdx.x;  // 262144 float4 total
  if (o < 262144) {
    const int bc = o >> 10;  // 1024 float4 per (b,c)
    const float* part = ws + WS_PART + bc * 4;
    const float s = ws[WS_C0] +
        (part[0] + part[1] + part[2] + part[3]) * (1.0f / (float)HW);
    out[o] = make_float4(s, s, s, s);
  }
}

extern "C" void kernel_launch(void* const* d_in, const int* in_sizes, int n_in,
                              void* d_out, int out_size, void* d_ws, size_t ws_size,
                              hipStream_t stream) {
  const float* x          = (const float*)d_in[0];   // [16,16,32,64,64]
  const float* conv_w     = (const float*)d_in[1];   // [32,1,3]
  const float* conv_b     = (const float*)d_in[2];   // [32]
  const float* scale_w    = (const float*)d_in[3];   // [4]
  const float* alpha_p    = (const float*)d_in[4];   // [1]
  const float* beta_p     = (const float*)d_in[5];   // [1]
  const float* t_imp      = (const float*)d_in[6];   // [32]
  // d_in[7] attn_key  : unused (softmax over singleton axis)
  const float* attn_value = (const float*)d_in[8];   // [4,8]
  // d_in[9] pos_encoding : unused (only feeds dead scores)
  const float* proj_w     = (const float*)d_in[10];  // [32,32]
  const float* proj_b     = (const float*)d_in[11];  // [32]
  const float* t_dyn      = (const float*)d_in[12];  // [32]
  const float* ln_gamma   = (const float*)d_in[13];  // [32]
  const float* ln_beta    = (const float*)d_in[14];  // [32]

  float* ws  = (float*)d_ws;
  float* out = (float*)d_out;

  etad_weights_kernel<<<1, 32, 0, stream>>>(
      conv_w, conv_b, scale_w, alpha_p, beta_p, t_imp, attn_value,
      proj_w, proj_b, t_dyn, ln_gamma, ln_beta, ws);

  dim3 gridR(256, 4);
  etad_reduce_kernel<<<gridR, 256, 0, stream>>>(x, ws);

  etad_broadcast_kernel<<<1024, 256, 0, stream>>>(ws, (float4*)out);
}